// E63mLiteCell_4887672783112
// MI455X (gfx1250) — compile-verified
//
#include <hip/hip_runtime.h>
#include <math.h>

typedef float v2f __attribute__((ext_vector_type(2)));
typedef float v8f __attribute__((ext_vector_type(8)));

#define T_STEPS 1024
#define BATCH   4
#define DDIM    512
#define NDIM    64
#define M_ROWS  (T_STEPS * BATCH)           // 4096 rows of x
#define NCOLS   (DDIM + DDIM + NDIM + NDIM) // 1152 concatenated output features

// ---------------------------------------------------------------------------
// Kernel 1: fused projection GEMM via fp32 WMMA (16x16x4).
// out[m, j] = sum_k x[m,k] * Wcat[j,k], j in {W_k | W_q | W_x | W_alpha}.
// Each wave computes a 16(M) x 64(N) strip: 4 accumulators sharing one A
// fragment per K-step -> 5 global b64 loads per 4 WMMA (vs 8 unblocked).
// All operands L2-resident (x: 8MB, weights: 2.3MB << 192MB L2).
// ---------------------------------------------------------------------------
__global__ __launch_bounds__(256)
void proj_gemm(const float* __restrict__ x,
               const float* __restrict__ Wk,
               const float* __restrict__ Wq,
               const float* __restrict__ Wx,
               const float* __restrict__ Wa,
               float* __restrict__ kbuf,
               float* __restrict__ qbuf,
               float* __restrict__ wxbuf,
               float* __restrict__ axbuf)
{
    const int tid  = threadIdx.x;
    const int wave = tid >> 5;
    const int lane = tid & 31;
    const int flat = blockIdx.x * 8 + wave;   // 4608 strips total
    const int mT   = flat & 255;              // 256 M-tiles
    const int nG   = flat >> 8;               // 18 N-groups of 64 columns
    const int m0   = mT * 16;
    const int n0   = nG * 64;
    const int l    = lane & 15;
    const int half = lane >> 4;

    // Select weight segment + output buffer (wave-uniform; 64-col groups never
    // straddle the 512|512|64|64 segment boundaries).
    const float* W; float* outp; int ostride; int ocol;
    if (n0 < 512)       { W = Wk + (size_t)n0 * DDIM;          outp = kbuf;  ostride = DDIM; ocol = n0; }
    else if (n0 < 1024) { W = Wq + (size_t)(n0 - 512) * DDIM;  outp = qbuf;  ostride = DDIM; ocol = n0 - 512; }
    else if (n0 < 1088) { W = Wx + (size_t)(n0 - 1024) * DDIM; outp = wxbuf; ostride = NDIM; ocol = n0 - 1024; }
    else                { W = Wa + (size_t)(n0 - 1088) * DDIM; outp = axbuf; ostride = NDIM; ocol = n0 - 1088; }

    // A fragment: 16x4 fp32. lanes 0-15: K={k0,k0+1}; lanes 16-31: K={k0+2,k0+3}.
    // B fragment i: 4x16 fp32, col n0+16i+l; B[k][n] = W[n*D + k] -> same
    // contiguous-pair pattern along k.
    const v2f* ap  = (const v2f*)(x + (size_t)(m0 + l) * DDIM + 2 * half);
    const v2f* bp0 = (const v2f*)(W + (size_t)(l     ) * DDIM + 2 * half);
    const v2f* bp1 = (const v2f*)(W + (size_t)(l + 16) * DDIM + 2 * half);
    const v2f* bp2 = (const v2f*)(W + (size_t)(l + 32) * DDIM + 2 * half);
    const v2f* bp3 = (const v2f*)(W + (size_t)(l + 48) * DDIM + 2 * half);

    v8f acc0 = {0.f,0.f,0.f,0.f,0.f,0.f,0.f,0.f};
    v8f acc1 = acc0, acc2 = acc0, acc3 = acc0;

    #pragma unroll 2
    for (int k0 = 0; k0 < DDIM; k0 += 4) {
        const int ki = k0 >> 1;
        v2f a  = ap[ki];
        v2f b0 = bp0[ki];
        v2f b1 = bp1[ki];
        v2f b2 = bp2[ki];
        v2f b3 = bp3[ki];
        acc0 = __builtin_amdgcn_wmma_f32_16x16x4_f32(false, a, false, b0, (short)0, acc0, false, false);
        acc1 = __builtin_amdgcn_wmma_f32_16x16x4_f32(false, a, false, b1, (short)0, acc1, false, false);
        acc2 = __builtin_amdgcn_wmma_f32_16x16x4_f32(false, a, false, b2, (short)0, acc2, false, false);
        acc3 = __builtin_amdgcn_wmma_f32_16x16x4_f32(false, a, false, b3, (short)0, acc3, false, false);
    }

    // C/D layout: VGPR r -> row m0+r (lanes 0-15) / m0+8+r (lanes 16-31),
    // col = tile_base + l.
    const int rbase = m0 + half * 8;
    float* o = outp + (size_t)rbase * ostride + ocol + l;
    #pragma unroll
    for (int r = 0; r < 8; r++) {
        const size_t ro = (size_t)r * ostride;
        o[ro     ] = acc0[r];
        o[ro + 16] = acc1[r];
        o[ro + 32] = acc2[r];
        o[ro + 48] = acc3[r];
    }
}

// ---------------------------------------------------------------------------
// Kernel 2: sequential scan. One 1024-thread workgroup per batch.
// State S (64x512 fp32 = 128 KB) lives in LDS (CDNA5: 320 KB/WGP).
// Thread (n,g): n = tid>>4 row, g = tid&15 d-group; each g owns float4s at
// index g + 16*j (j=0..7) -> 16 consecutive lanes span all 64 LDS banks.
// 16-way reductions over g use wave32 __shfl_xor (lane halves reduce
// independently; masks 1,2,4,8 never cross the 16-lane boundary).
// ---------------------------------------------------------------------------
__global__ __launch_bounds__(1024)
void scan_kernel(const float* __restrict__ kbuf,
                 const float* __restrict__ qbuf,
                 const float* __restrict__ wxbuf,
                 const float* __restrict__ axbuf,
                 const float* __restrict__ Wr,
                 const float* __restrict__ bvec,
                 const float* __restrict__ balpha,
                 const float* __restrict__ Wout,
                 const float* __restrict__ S0,
                 float* __restrict__ ys,
                 float* __restrict__ Sfull)
{
    const int b   = blockIdx.x;
    const int tid = threadIdx.x;

    __shared__ __align__(16) float sS[NDIM * DDIM];  // 128 KB state
    __shared__ __align__(16) float skq[2 * DDIM];    // k_t | q_t
    __shared__ __align__(16) float sWxAx[2 * NDIM];  // wx_t | ax_t
    __shared__ __align__(16) float sRed[NDIM];       // Sk / Sq reduction slot
    __shared__ __align__(16) float sTk[NDIM];        // tanh(Sk)
    __shared__ __align__(16) float sA[NDIM];         // sigmoid gate a
    __shared__ __align__(16) float sC[NDIM];         // (1-a)*v
    __shared__ __align__(16) float sTq[NDIM];        // tanh(Sq)

    // Prologue: S <- S0 (LDS) and S_full[0] <- S0.
    {
        const float4* s04 = (const float4*)(S0 + (size_t)b * NDIM * DDIM);
        float4* ss4 = (float4*)sS;
        float4* so4 = (float4*)(Sfull + (size_t)b * NDIM * DDIM);
        for (int i = tid; i < NDIM * DDIM / 4; i += 1024) {
            float4 v = s04[i];
            ss4[i] = v;
            so4[i] = v;
        }
    }

    const int n = tid >> 4;
    const int g = tid & 15;

    for (int t = 0; t < T_STEPS; t++) {
        __syncthreads();
        // ---- Phase 0: stage k_t, q_t, wx_t, ax_t into LDS -----------------
        {
            const size_t base = (size_t)t * BATCH + b;
            if (tid < DDIM) skq[tid] = kbuf[base * DDIM + tid];
            else            skq[tid] = qbuf[base * DDIM + (tid - DDIM)];
            if (tid < NDIM)          sWxAx[tid] = wxbuf[base * NDIM + tid];
            else if (tid < 2 * NDIM) sWxAx[tid] = axbuf[base * NDIM + (tid - NDIM)];
        }
        __syncthreads();

        const float4* k4 = (const float4*)skq;
        const float4* q4 = (const float4*)(skq + DDIM);

        // ---- Phase 1: Sk[n] = S[n,:] . k ---------------------------------
        float part = 0.f;
        {
            const float4* srow = (const float4*)(sS + (size_t)n * DDIM);
            #pragma unroll
            for (int j = 0; j < 8; j++) {
                const int idx = g + (j << 4);
                float4 s = srow[idx], kk = k4[idx];
                part += s.x * kk.x + s.y * kk.y + s.z * kk.z + s.w * kk.w;
            }
        }
        part += __shfl_xor(part, 1);
        part += __shfl_xor(part, 2);
        part += __shfl_xor(part, 4);
        part += __shfl_xor(part, 8);
        if (g == 0) sRed[n] = part;
        __syncthreads();

        // ---- Phase 2: tanh(Sk) and gate a ---------------------------------
        if (tid < NDIM) {
            sTk[tid] = tanhf(sRed[tid]);
            const float ax = sWxAx[NDIM + tid] + balpha[tid];
            sA[tid] = 1.f / (1.f + expf(-ax));
        }
        __syncthreads();

        // ---- Phase 3: v = tanh(tanh(Sk) @ W_r^T + wx + b); c = (1-a)*v ----
        if (tid < NDIM) {
            float acc = sWxAx[tid] + bvec[tid];
            const float4* wr4 = (const float4*)(Wr + (size_t)tid * NDIM); // L2-resident
            const float4* tk4 = (const float4*)sTk;
            #pragma unroll
            for (int m = 0; m < NDIM / 4; m++) {
                float4 w = wr4[m], tk = tk4[m];
                acc += w.x * tk.x + w.y * tk.y + w.z * tk.z + w.w * tk.w;
            }
            sC[tid] = (1.f - sA[tid]) * tanhf(acc);
        }
        __syncthreads();

        // ---- Phase 4: S' = a*S + c*k (fused with Sq = S'.q and HBM store) --
        float qpart = 0.f;
        {
            const float an = sA[n];
            const float cn = sC[n];
            float4* srow = (float4*)(sS + (size_t)n * DDIM);
            float4* gout = (float4*)(Sfull +
                (((size_t)(t + 1) * BATCH + b) * NDIM + n) * DDIM);
            #pragma unroll
            for (int j = 0; j < 8; j++) {
                const int idx = g + (j << 4);
                float4 s = srow[idx], kk = k4[idx], qq = q4[idx];
                s.x = an * s.x + cn * kk.x;
                s.y = an * s.y + cn * kk.y;
                s.z = an * s.z + cn * kk.z;
                s.w = an * s.w + cn * kk.w;
                srow[idx] = s;
                gout[idx] = s;   // S_full[t+1] -> HBM (fire-and-forget store)
                qpart += s.x * qq.x + s.y * qq.y + s.z * qq.z + s.w * qq.w;
            }
        }
        qpart += __shfl_xor(qpart, 1);
        qpart += __shfl_xor(qpart, 2);
        qpart += __shfl_xor(qpart, 4);
        qpart += __shfl_xor(qpart, 8);
        if (g == 0) sRed[n] = qpart;
        __syncthreads();

        if (tid < NDIM) sTq[tid] = tanhf(sRed[tid]);
        __syncthreads();

        // ---- Phase 5: y[d] = sum_n tanh(Sq)[n] * W_out[d,n] ---------------
        if (tid < DDIM) {
            const float4* w4  = (const float4*)(Wout + (size_t)tid * NDIM); // L2-resident
            const float4* tq4 = (const float4*)sTq;
            float acc = 0.f;
            #pragma unroll
            for (int m = 0; m < NDIM / 4; m++) {
                float4 w = w4[m], tv = tq4[m];
                acc += w.x * tv.x + w.y * tv.y + w.z * tv.z + w.w * tv.w;
            }
            ys[((size_t)t * BATCH + b) * DDIM + tid] = acc;
        }
    }
}

// ---------------------------------------------------------------------------
extern "C" void kernel_launch(void* const* d_in, const int* in_sizes, int n_in,
                              void* d_out, int out_size, void* d_ws, size_t ws_size,
                              hipStream_t stream)
{
    (void)in_sizes; (void)n_in; (void)out_size; (void)ws_size;

    const float* x      = (const float*)d_in[0];
    const float* Wk     = (const float*)d_in[1];
    const float* Wq     = (const float*)d_in[2];
    const float* Wx     = (const float*)d_in[3];
    const float* Wr     = (const float*)d_in[4];
    const float* bvec   = (const float*)d_in[5];
    const float* Wa     = (const float*)d_in[6];
    const float* balpha = (const float*)d_in[7];
    const float* Wout   = (const float*)d_in[8];
    const float* S0     = (const float*)d_in[9];

    // Workspace layout: k_all | q_all | wx_all | ax_all  (18 MB fp32)
    float* ws    = (float*)d_ws;
    float* kbuf  = ws;
    float* qbuf  = kbuf  + (size_t)M_ROWS * DDIM;
    float* wxbuf = qbuf  + (size_t)M_ROWS * DDIM;
    float* axbuf = wxbuf + (size_t)M_ROWS * NDIM;

    // Output layout: ys (T,B,D) then S_full (T+1,B,N,D), fp32.
    float* ys    = (float*)d_out;
    float* Sfull = ys + (size_t)T_STEPS * BATCH * DDIM;

    // 256 M-tiles x 18 N-groups = 4608 waves, 8 waves per block -> 576 blocks.
    proj_gemm<<<576, 256, 0, stream>>>(x, Wk, Wq, Wx, Wa,
                                       kbuf, qbuf, wxbuf, axbuf);

    // One workgroup per batch; S state lives in 128 KB of LDS.
    scan_kernel<<<BATCH, 1024, 0, stream>>>(kbuf, qbuf, wxbuf, axbuf,
                                            Wr, bvec, balpha, Wout, S0,
                                            ys, Sfull);
}